// NonMaxSupression_42039139893803
// MI455X (gfx1250) — compile-verified
//
#include <hip/hip_runtime.h>

// Problem constants (match reference).
#define BN   16
#define AN   4096
#define KN   2048
#define PN   200
// IOU_THR = 0.5 exactly =>  iou > 0.5  <=>  3*inter > (area_i + area_j), with union > 0.

typedef __attribute__((ext_vector_type(2))) float v2f;
typedef __attribute__((ext_vector_type(8))) float v8f;

// ---------------------------------------------------------------------------
// Kernel 1: per-batch bitonic top-K sort (descending conf, ascending idx ties)
// One 1024-thread workgroup per batch; 32 KB LDS of (key, idx).
// Emits sorted conf, gathered boxes (float4 y1,x1,y2,x2) and precomputed areas.
// ---------------------------------------------------------------------------
__global__ __launch_bounds__(1024) void topk_sort_kernel(
    const float* __restrict__ conf, const float4* __restrict__ boxes,
    float* __restrict__ sconf, float4* __restrict__ sboxes,
    float* __restrict__ sarea) {
  const int b = blockIdx.x;
  const int tid = threadIdx.x;
  __shared__ float key[AN];
  __shared__ int   idx[AN];

  for (int i = tid; i < AN; i += 1024) { key[i] = conf[b * AN + i]; idx[i] = i; }
  __syncthreads();

  for (int k = 2; k <= AN; k <<= 1) {
    for (int j = k >> 1; j > 0; j >>= 1) {
      for (int i = tid; i < AN; i += 1024) {
        int p = i ^ j;
        if (p > i) {
          float ka = key[i], kb = key[p];
          int   ia = idx[i], ib = idx[p];
          // "i-element belongs before p-element" in descending-conf order
          bool beforeIP = (ka > kb) || (ka == kb && ia < ib);
          bool dirDesc  = ((i & k) == 0);
          bool doSwap   = dirDesc ? !beforeIP : beforeIP;
          if (doSwap) { key[i] = kb; key[p] = ka; idx[i] = ib; idx[p] = ia; }
        }
      }
      __syncthreads();
    }
  }

  for (int r = tid; r < KN; r += 1024) {
    float cv = key[r];
    int   id = idx[r];
    float4 bx = boxes[b * AN + id];            // (y1, x1, y2, x2)
    sconf [b * KN + r] = cv;
    sboxes[b * KN + r] = bx;
    sarea [b * KN + r] = (bx.z - bx.x) * (bx.w - bx.y);
  }
}

// ---------------------------------------------------------------------------
// Kernel 2: suppression bitmask. One wave32 computes a 16-row x 32-col tile.
// S[i][j] = area_i + area_j is produced by V_WMMA_F32_16X16X4_F32 as the
// rank-2 product [area_i, 1] x [1 ; area_j]; intersection is VALU min/max;
// __ballot transposes lane predicates into per-row 32-bit mask words.
// Block = 256 threads (8 waves) sharing the row tile staged in LDS.
// mask[b][i][jw] bit (j%32) set  <=>  iou(i, jw*32 + j%32) > 0.5
// Tiles strictly below the diagonal are skipped (never read by kernel 3).
// ---------------------------------------------------------------------------
__global__ __launch_bounds__(256) void iou_mask_kernel(
    const float4* __restrict__ boxes, const float* __restrict__ area,
    unsigned* __restrict__ mask) {
  const int lane = threadIdx.x & 31;
  const int wave = threadIdx.x >> 5;
  const int it   = blockIdx.x;                 // row tile (16 rows)
  const int b    = blockIdx.z;
  const int i0   = it * 16;
  const int jt   = blockIdx.y * 8 + wave;      // col tile (32 cols)
  const int j0   = jt * 32;

  __shared__ float4 srow[16];
  __shared__ float  sar [16];
  __shared__ float4 scol[8][32];
  __shared__ float  sac [8][32];

  if (threadIdx.x < 16) {
    srow[threadIdx.x] = boxes[b * KN + i0 + threadIdx.x];
    sar [threadIdx.x] = area [b * KN + i0 + threadIdx.x];
  }
  scol[wave][lane] = boxes[b * KN + j0 + lane];
  sac [wave][lane] = area [b * KN + j0 + lane];
  __syncthreads();

  if (j0 + 31 < i0) return;                    // wave-uniform: below diagonal

  // WMMA operands (wave32 layout: lanes 0-15 hold K=0/1, lanes 16-31 K=2/3).
  const bool lo = lane < 16;
  const int  ln = lane & 15;
  v2f a, bA, bB;
  v8f cz = {};
  a.x  = lo ? sar[ln]            : 0.0f;       // A col K=0: area_i
  a.y  = lo ? 1.0f               : 0.0f;       // A col K=1: ones
  bA.x = lo ? 1.0f               : 0.0f;       // B row K=0: ones
  bA.y = lo ? sac[wave][ln]      : 0.0f;       // B row K=1: area_j (cols 0..15)
  bB.x = bA.x;
  bB.y = lo ? sac[wave][16 + ln] : 0.0f;       // cols 16..31

  v8f SA = __builtin_amdgcn_wmma_f32_16x16x4_f32(false, a, false, bA,
                                                 (short)0, cz, false, false);
  v8f SB = __builtin_amdgcn_wmma_f32_16x16x4_f32(false, a, false, bB,
                                                 (short)0, cz, false, false);

  const float4 cbA = scol[wave][ln];
  const float4 cbB = scol[wave][16 + ln];
  const int rbase = lo ? 0 : 8;                // C/D: lanes>=16 hold rows M+8

  unsigned balA[8], balB[8];
#pragma unroll
  for (int r = 0; r < 8; ++r) {
    float4 rb = srow[rbase + r];
    // tile A (cols j0 .. j0+15)
    float ihA = fminf(rb.z, cbA.z) - fmaxf(rb.x, cbA.x);
    float iwA = fminf(rb.w, cbA.w) - fmaxf(rb.y, cbA.y);
    float inA = fmaxf(ihA, 0.0f) * fmaxf(iwA, 0.0f);
    float sA  = SA[r];
    bool  pA  = (3.0f * inA > sA) && (sA - inA > 0.0f);
    balA[r]   = (unsigned)__ballot(pA);
    // tile B (cols j0+16 .. j0+31)
    float ihB = fminf(rb.z, cbB.z) - fmaxf(rb.x, cbB.x);
    float iwB = fminf(rb.w, cbB.w) - fmaxf(rb.y, cbB.y);
    float inB = fmaxf(ihB, 0.0f) * fmaxf(iwB, 0.0f);
    float sB  = SB[r];
    bool  pB  = (3.0f * inB > sB) && (sB - inB > 0.0f);
    balB[r]   = (unsigned)__ballot(pB);
  }

  // Assemble per-row 32-bit words (ballot low 16 bits = rows M, high = M+8).
  unsigned rw[16];
#pragma unroll
  for (int r = 0; r < 8; ++r) {
    rw[r]     = (balA[r] & 0xFFFFu) | ((balB[r] & 0xFFFFu) << 16);
    rw[r + 8] = (balA[r] >> 16)     | (balB[r] & 0xFFFF0000u);
  }
  if (lane < 16) {
    unsigned w = rw[0];
#pragma unroll
    for (int t = 1; t < 16; ++t) w = (lane == t) ? rw[t] : w;
    mask[((size_t)b * KN + i0 + lane) * 64 + jt] = w;
  }
}

// ---------------------------------------------------------------------------
// Kernel 3: per-batch greedy reduce + output gather. One wave32 per batch.
// Keep vector: 2048 bits = 64 bits/lane in registers; shuffle-broadcast the
// owner word each step (no barriers). Then compact kept indices via a wave
// prefix sum into LDS and emit the reference's P=200 gather with padding.
// ---------------------------------------------------------------------------
__device__ inline unsigned long long bcast64(unsigned long long v, int src) {
  unsigned int lo = __shfl((unsigned int)v, src, 32);
  unsigned int hi = __shfl((unsigned int)(v >> 32), src, 32);
  return ((unsigned long long)hi << 32) | lo;
}

__global__ __launch_bounds__(32) void nms_finalize_kernel(
    const unsigned* __restrict__ mask, const float* __restrict__ sconf,
    const float4* __restrict__ sboxes, float* __restrict__ out) {
  const int b = blockIdx.x;
  const int lane = threadIdx.x;
  const unsigned long long* mrow =
      (const unsigned long long*)(mask + (size_t)b * KN * 64);

  unsigned long long kw = ~0ull;               // lane owns j in [64*lane, 64*lane+63]
  for (int i = 0; i < KN; ++i) {
    const int owner = i >> 6;
    const unsigned long long w = bcast64(kw, owner);
    const unsigned long long* rp = mrow + (size_t)i * 32;
    __builtin_prefetch(rp + 32 + lane, 0, 0);  // next row -> global_prefetch_b8
    if ((w >> (i & 63)) & 1ull) {              // keep[i] -> suppress later j
      unsigned long long m = 0ull;
      if (lane >= owner) {
        m = rp[lane];
        if (lane == owner) {
          int bit = i & 63;                    // clear j <= i
          unsigned long long low = (bit == 63) ? ~0ull
                                               : ((1ull << (bit + 1)) - 1ull);
          m &= ~low;
        }
      }
      kw &= ~m;
    }
  }

  // Wave-exclusive prefix of per-lane popcounts.
  int pc = __popcll(kw);
  int x = pc;
  for (int off = 1; off < 32; off <<= 1) {
    int y = __shfl_up(x, off);
    if (lane >= off) x += y;
  }
  const int excl  = x - pc;
  const int total = __shfl(x, 31);

  __shared__ int order[KN];
  {
    unsigned long long t = kw;
    int r = excl;
    while (t) {
      int bit = __builtin_ctzll(t);
      order[r++] = lane * 64 + bit;
      t &= t - 1ull;
    }
  }
  __syncthreads();

  const int n          = total < PN ? total : PN;
  const int last_idx   = (n > 0) ? order[n - 1] : 0;
  const int additional = PN - n;
  const int s_cap      = KN - additional;
  const int s_nxt      = last_idx + 1;
  const int start      = s_nxt < s_cap ? s_nxt : s_cap;

  const float*  cfb = sconf  + b * KN;
  const float4* bxb = sboxes + b * KN;
  float*  oc = out + b * PN;                         // conf_out (B,P)
  float4* ob = (float4*)(out + BN * PN) + b * PN;    // box_out  (B,P,4)
  for (int p = lane; p < PN; p += 32) {
    int g = (p < n) ? order[p] : (start + (p - n));
    oc[p] = cfb[g];
    ob[p] = bxb[g];
  }
}

// ---------------------------------------------------------------------------
// Launch. Workspace layout (floats):
//   [0)       sconf   B*K           =  32768 f (128 KB)
//   [32768)   sboxes  B*K*4         = 131072 f (512 KB)
//   [163840)  sarea   B*K           =  32768 f (128 KB)
//   [196608)  mask    B*K*64 words  = 8 MB
// Total ~9.2 MB.
// ---------------------------------------------------------------------------
extern "C" void kernel_launch(void* const* d_in, const int* in_sizes, int n_in,
                              void* d_out, int out_size, void* d_ws,
                              size_t ws_size, hipStream_t stream) {
  (void)in_sizes; (void)n_in; (void)out_size; (void)ws_size;
  const float* conf  = (const float*)d_in[0];
  const float* boxes = (const float*)d_in[1];

  float*    ws     = (float*)d_ws;
  float*    sconf  = ws;
  float4*   sboxes = (float4*)(ws + 32768);
  float*    sarea  = ws + 163840;
  unsigned* maskw  = (unsigned*)(ws + 196608);

  topk_sort_kernel<<<BN, 1024, 0, stream>>>(conf, (const float4*)boxes,
                                            sconf, sboxes, sarea);
  iou_mask_kernel<<<dim3(KN / 16, 8, BN), 256, 0, stream>>>(
      (const float4*)sboxes, sarea, maskw);
  nms_finalize_kernel<<<BN, 32, 0, stream>>>(maskw, sconf,
                                             (const float4*)sboxes,
                                             (float*)d_out);
}